// RunningExpectedCalibrationError_26096221290826
// MI455X (gfx1250) — compile-verified
//
#include <hip/hip_runtime.h>
#include <hip/hip_bf16.h>
#include <stdint.h>

// ---------------------------------------------------------------------------
// ECE on MI455X (gfx1250): HBM-bandwidth-bound rowwise softmax-max reduction.
// 512 MB one-pass stream -> ~22 us roofline at 23.3 TB/s. Rows are staged
// HBM -> LDS with CDNA5 async loads (global_load_async_to_lds_b128, NT hint,
// ASYNCcnt), pipelined 8 deep per wave so that resident waves hold >16 MB in
// flight (bandwidth-delay product at ~700 ns HBM latency).
// ---------------------------------------------------------------------------

#define WAVES_PER_BLOCK 8
#define BLOCK_THREADS   (WAVES_PER_BLOCK * 32)
#define GRID_BLOCKS     2048
#define DEPTH           8          // async pipeline depth (rows in flight/wave)

// Issue one async 16B-per-lane copy: 32 lanes x 16B = one 512B row -> LDS.
// Non-temporal: the logit stream (512MB) is read once and exceeds the 192MB L2.
__device__ __forceinline__ void async_row_to_lds(uint32_t lds_byte_off,
                                                 const float* gaddr) {
  asm volatile("global_load_async_to_lds_b128 %0, %1, off th:TH_LOAD_NT"
               :: "v"(lds_byte_off), "v"(gaddr)
               : "memory");
}

__device__ __forceinline__ void wait_async_le7() {
  asm volatile("s_wait_asynccnt 7" ::: "memory");
}
__device__ __forceinline__ void wait_async_le0() {
  asm volatile("s_wait_asynccnt 0" ::: "memory");
}

// ---------------------------------------------------------------------------
// Kernel 1 (C == 128 fast path): one wave per row, DEPTH-deep async pipeline.
// LDS layout (bytes): wave w: [w*DEPTH*512, (w+1)*DEPTH*512) = DEPTH x 512B.
// After the row region: WAVES_PER_BLOCK*2 floats of block-reduce scratch.
// ---------------------------------------------------------------------------
__global__ void ece_rows_c128(const float* __restrict__ logits,
                              const int* __restrict__ target,
                              float* __restrict__ partials,
                              int nrows) {
  extern __shared__ __align__(16) float smem[];

  const int lane = threadIdx.x & 31;
  const int wib  = threadIdx.x >> 5;                         // wave in block
  const int wavesTotal = gridDim.x * WAVES_PER_BLOCK;
  const uint32_t ldsWaveBase = (uint32_t)wib * (DEPTH * 512u);  // bytes
  const uint32_t ldsLane     = (uint32_t)lane * 16u;            // bytes

  float accT = 0.0f, confT = 0.0f;

  const int r0 = blockIdx.x * WAVES_PER_BLOCK + wib;         // wave-uniform
  if (r0 < nrows) {
    // --- prologue: fill the pipeline (up to DEPTH rows in flight) ---------
    int rIssue = r0;
    #pragma unroll
    for (int d = 0; d < DEPTH; ++d) {
      if (rIssue < nrows) {
        async_row_to_lds(ldsWaveBase + (uint32_t)d * 512u + ldsLane,
                         logits + (size_t)rIssue * 128 + lane * 4);
        rIssue += wavesTotal;
      }
    }

    int rProc = r0;
    int buf   = 0;

    // --- steady state: exactly DEPTH in flight at each wait ---------------
    while (rIssue < nrows) {
      wait_async_le7();                       // oldest row has landed

      const float4 v =
          reinterpret_cast<const float4*>(smem)[wib * (DEPTH * 32) +
                                                buf * 32 + lane];

      // Local max + argmax (first occurrence on ties: strict >).
      float m  = v.x; int id = lane * 4;
      if (v.y > m) { m = v.y; id = lane * 4 + 1; }
      if (v.z > m) { m = v.z; id = lane * 4 + 2; }
      if (v.w > m) { m = v.w; id = lane * 4 + 3; }
      #pragma unroll
      for (int off = 16; off > 0; off >>= 1) {
        const float om = __shfl_xor(m, off, 32);
        const int   oi = __shfl_xor(id, off, 32);
        if (om > m || (om == m && oi < id)) { m = om; id = oi; }
      }

      // Sum of exp(x - max); top-1 prob = 1/sum (exp(0)=1 term included).
      float s = __expf(v.x - m) + __expf(v.y - m) +
                __expf(v.z - m) + __expf(v.w - m);
      #pragma unroll
      for (int off = 16; off > 0; off >>= 1) s += __shfl_xor(s, off, 32);

      const int t = target[rProc];            // wave-uniform -> scalar load
      accT  += (id == t) ? 1.0f : 0.0f;
      confT += 1.0f / s;

      // Refill the slot we just consumed.
      async_row_to_lds(ldsWaveBase + (uint32_t)buf * 512u + ldsLane,
                       logits + (size_t)rIssue * 128 + lane * 4);
      rIssue += wavesTotal;
      rProc  += wavesTotal;
      buf = (buf + 1) & (DEPTH - 1);
    }

    // --- drain: <= DEPTH buffered rows remain; wait for all of them -------
    wait_async_le0();
    while (rProc < nrows) {
      const float4 v =
          reinterpret_cast<const float4*>(smem)[wib * (DEPTH * 32) +
                                                buf * 32 + lane];
      float m  = v.x; int id = lane * 4;
      if (v.y > m) { m = v.y; id = lane * 4 + 1; }
      if (v.z > m) { m = v.z; id = lane * 4 + 2; }
      if (v.w > m) { m = v.w; id = lane * 4 + 3; }
      #pragma unroll
      for (int off = 16; off > 0; off >>= 1) {
        const float om = __shfl_xor(m, off, 32);
        const int   oi = __shfl_xor(id, off, 32);
        if (om > m || (om == m && oi < id)) { m = om; id = oi; }
      }
      float s = __expf(v.x - m) + __expf(v.y - m) +
                __expf(v.z - m) + __expf(v.w - m);
      #pragma unroll
      for (int off = 16; off > 0; off >>= 1) s += __shfl_xor(s, off, 32);

      const int t = target[rProc];
      accT  += (id == t) ? 1.0f : 0.0f;
      confT += 1.0f / s;

      rProc += wavesTotal;
      buf = (buf + 1) & (DEPTH - 1);
    }
  }

  // Block reduce (fixed order -> deterministic), one partial pair per block.
  float* red = smem + WAVES_PER_BLOCK * DEPTH * 128;
  if (lane == 0) { red[wib * 2] = accT; red[wib * 2 + 1] = confT; }
  __syncthreads();
  if (threadIdx.x == 0) {
    float a = 0.0f, c = 0.0f;
    #pragma unroll
    for (int w = 0; w < WAVES_PER_BLOCK; ++w) {
      a += red[w * 2];
      c += red[w * 2 + 1];
    }
    partials[(size_t)blockIdx.x * 2]     = a;
    partials[(size_t)blockIdx.x * 2 + 1] = c;
  }
}

// ---------------------------------------------------------------------------
// Generic fallback (any C): one wave per row, direct global loads, two passes.
// ---------------------------------------------------------------------------
__global__ void ece_rows_generic(const float* __restrict__ logits,
                                 const int* __restrict__ target,
                                 float* __restrict__ partials,
                                 int nrows, int C) {
  __shared__ float red[WAVES_PER_BLOCK * 2];
  const int lane = threadIdx.x & 31;
  const int wib  = threadIdx.x >> 5;
  const int wavesTotal = gridDim.x * WAVES_PER_BLOCK;

  float accT = 0.0f, confT = 0.0f;
  for (int r = blockIdx.x * WAVES_PER_BLOCK + wib; r < nrows; r += wavesTotal) {
    const float* row = logits + (size_t)r * C;
    float m = -3.402823466e38f; int id = 0x7fffffff;
    for (int c = lane; c < C; c += 32) {
      const float x = row[c];
      if (x > m) { m = x; id = c; }       // c increasing -> first occurrence
    }
    #pragma unroll
    for (int off = 16; off > 0; off >>= 1) {
      const float om = __shfl_xor(m, off, 32);
      const int   oi = __shfl_xor(id, off, 32);
      if (om > m || (om == m && oi < id)) { m = om; id = oi; }
    }
    float s = 0.0f;
    for (int c = lane; c < C; c += 32) s += __expf(row[c] - m);
    #pragma unroll
    for (int off = 16; off > 0; off >>= 1) s += __shfl_xor(s, off, 32);

    const int t = target[r];
    accT  += (id == t) ? 1.0f : 0.0f;
    confT += 1.0f / s;
  }

  if (lane == 0) { red[wib * 2] = accT; red[wib * 2 + 1] = confT; }
  __syncthreads();
  if (threadIdx.x == 0) {
    float a = 0.0f, c = 0.0f;
    #pragma unroll
    for (int w = 0; w < WAVES_PER_BLOCK; ++w) {
      a += red[w * 2];
      c += red[w * 2 + 1];
    }
    partials[(size_t)blockIdx.x * 2]     = a;
    partials[(size_t)blockIdx.x * 2 + 1] = c;
  }
}

// ---------------------------------------------------------------------------
// Kernel 2: deterministic fixed-order reduction of block partials.
// ECE = |sum(acc) - sum(conf)| / N   (prop/N == 1, bins cancel).
// ---------------------------------------------------------------------------
__global__ void ece_finalize(const float* __restrict__ partials,
                             float* __restrict__ out,
                             int nblocks, float invN) {
  __shared__ float sa[256];
  __shared__ float sc[256];
  float a = 0.0f, c = 0.0f;
  for (int i = threadIdx.x; i < nblocks; i += 256) {
    a += partials[(size_t)i * 2];
    c += partials[(size_t)i * 2 + 1];
  }
  sa[threadIdx.x] = a;
  sc[threadIdx.x] = c;
  __syncthreads();
  for (int s = 128; s > 0; s >>= 1) {
    if (threadIdx.x < s) {
      sa[threadIdx.x] += sa[threadIdx.x + s];
      sc[threadIdx.x] += sc[threadIdx.x + s];
    }
    __syncthreads();
  }
  if (threadIdx.x == 0) out[0] = fabsf(sa[0] - sc[0]) * invN;
}

// ---------------------------------------------------------------------------
extern "C" void kernel_launch(void* const* d_in, const int* in_sizes, int n_in,
                              void* d_out, int out_size, void* d_ws, size_t ws_size,
                              hipStream_t stream) {
  const float* logits = (const float*)d_in[0];
  const int*   target = (const int*)d_in[1];
  float*       out    = (float*)d_out;
  float*       parts  = (float*)d_ws;

  const int nrows = in_sizes[1];
  const int C     = (nrows > 0) ? (in_sizes[0] / nrows) : 0;

  int blocks = GRID_BLOCKS;
  const size_t needed = (size_t)blocks * 2 * sizeof(float);
  if (ws_size < needed) {
    blocks = (int)(ws_size / (2 * sizeof(float)));
    if (blocks < 1) blocks = 1;
  }

  if (C == 128) {
    const size_t shmem =
        (WAVES_PER_BLOCK * DEPTH * 128 + WAVES_PER_BLOCK * 2) * sizeof(float);
    ece_rows_c128<<<dim3(blocks), dim3(BLOCK_THREADS), shmem, stream>>>(
        logits, target, parts, nrows);
  } else {
    ece_rows_generic<<<dim3(blocks), dim3(BLOCK_THREADS), 0, stream>>>(
        logits, target, parts, nrows, C);
  }
  ece_finalize<<<dim3(1), dim3(256), 0, stream>>>(
      parts, out, blocks, 1.0f / (float)(nrows > 0 ? nrows : 1));
}